// UpBlock_30915174596690
// MI455X (gfx1250) — compile-verified
//
#include <hip/hip_runtime.h>
#include <cstdint>
#include <cstddef>

typedef __attribute__((ext_vector_type(2))) float v2f;
typedef __attribute__((ext_vector_type(8))) float v8f;
typedef long long i64;

#define DEVFN static __device__ __forceinline__

DEVFN float siluf(float x) { return x / (1.f + __expf(-x)); }
DEVFN float geluf(float x) {
  float x3 = x * x * x;
  return 0.5f * x * (1.f + tanhf(0.7978845608028654f * (x + 0.044715f * x3)));
}
DEVFN float softplusf(float x) { return x > 20.f ? x : log1pf(__expf(x)); }

// ---------------------------------------------------------------------------
// Generic strided batched GEMM via V_WMMA_F32_16X16X4_F32.
// C(m,n) = act( sum_k A(m,k)*B(n,k) + bias[n] ) + R(m,n)
// z = blockIdx.z decomposed as zb = z/zdiv, zk = z%zdiv (for (batch,dir) grids).
// A tile (16 x K) staged in LDS, shared by the 8 waves (one 16-wide N tile each).
// ---------------------------------------------------------------------------
__launch_bounds__(256)
__global__ void gemm_wmma(
    const float* __restrict__ A, const float* __restrict__ Bw,
    const float* __restrict__ bias, const float* __restrict__ R,
    float* __restrict__ C,
    int M, int N, int K, int act, int zdiv,
    i64 A_zb, i64 A_zk, i64 As_m, i64 As_k,
    i64 B_zk, i64 Bs_n, i64 Bs_k,
    i64 C_zb, i64 C_zk, i64 Cs_m, i64 Cs_n,
    i64 bias_zk, i64 R_zb, i64 R_zk)
{
  __shared__ float Alds[16 * 512];   // K <= 512 in all uses
  int z = blockIdx.z;
  int zb = z / zdiv, zk = z - zb * zdiv;
  int m0 = blockIdx.x * 16;
  const float* Az = A + zb * A_zb + zk * A_zk;
  int tot = 16 * K;
  for (int idx = threadIdx.x; idx < tot; idx += 256) {
    int m = idx / K, kk = idx - m * K;
    Alds[idx] = Az[(i64)(m0 + m) * As_m + (i64)kk * As_k];
  }
  __syncthreads();
  int wave = threadIdx.x >> 5, lane = threadIdx.x & 31;
  int lm = lane & 15, hi = lane >> 4;
  int n0 = (blockIdx.y * 8 + wave) * 16;
  if (n0 >= N) return;                    // wave-uniform exit (EXEC all-1 for WMMA)
  int nc = n0 + lm;
  int nl = nc < N ? nc : N - 1;           // clamp loads for ragged N (e.g. N=40)
  const float* Bb = Bw + zk * B_zk + (i64)nl * Bs_n;
  const float* Arow = Alds + lm * K;
  v8f acc = {};
  for (int k0 = 0; k0 < K; k0 += 4) {
    int ka = k0 + 2 * hi;
    v2f a, b;
    a[0] = Arow[ka];
    a[1] = Arow[ka + 1];
    b[0] = Bb[(i64)ka * Bs_k];
    b[1] = Bb[(i64)(ka + 1) * Bs_k];
    acc = __builtin_amdgcn_wmma_f32_16x16x4_f32(false, a, false, b, (short)0,
                                                acc, false, false);
  }
  if (nc >= N) return;
  float bv = bias ? bias[zk * bias_zk + nc] : 0.f;
  float* Cz = C + zb * C_zb + zk * C_zk;
  const float* Rz = R ? (R + zb * R_zb + zk * R_zk) : nullptr;
#pragma unroll
  for (int r = 0; r < 8; ++r) {
    int mm = m0 + r + 8 * hi;
    float v = acc[r] + bv;
    if (act == 1) v = siluf(v);
    else if (act == 2) v = geluf(v);
    else if (act == 3) v = softplusf(v);
    i64 off = (i64)mm * Cs_m + (i64)nc * Cs_n;
    if (Rz) v += Rz[off];
    Cz[off] = v;
  }
}

// ---------------------------------------------------------------------------
// 3x3 SAME conv (NCHW) as implicit GEMM with WMMA. Block: 16 output pixels
// (one row segment) x up to 128 output channels (8 waves x 16). Input patch
// (Cin x 3 x 18) staged in LDS once and reused by all waves. Weight stream is
// unit-stride per lane (k = ci*9 + j). Optional residual add (NCHW).
// ---------------------------------------------------------------------------
__launch_bounds__(256)
__global__ void conv3x3_wmma(
    const float* __restrict__ in, const float* __restrict__ w,
    const float* __restrict__ bias, const float* __restrict__ R,
    float* __restrict__ out,
    int Cin, int Cout, int H, int W,
    i64 in_bz, i64 out_bz, i64 R_bz)
{
  __shared__ float patch[13824];        // up to 256 * 54 floats = 54 KB
  int b = blockIdx.z;
  int segs = W >> 4;
  int y = blockIdx.x / segs;
  int x0 = (blockIdx.x - y * segs) << 4;
  int L = H * W;
  const float* inb = in + (i64)b * in_bz;
  int tot = Cin * 54;
  for (int idx = threadIdx.x; idx < tot; idx += 256) {
    int ci = idx / 54;
    int rem = idx - ci * 54;
    int ry = rem / 18, rx = rem - ry * 18;
    int gy = y + ry - 1, gx = x0 + rx - 1;
    float v = 0.f;
    if (gy >= 0 && gy < H && gx >= 0 && gx < W)
      v = inb[(i64)ci * L + gy * W + gx];
    patch[idx] = v;
  }
  __syncthreads();
  int wave = threadIdx.x >> 5, lane = threadIdx.x & 31;
  int lm = lane & 15, hi = lane >> 4;
  int n0 = (blockIdx.y * 8 + wave) * 16;
  if (n0 >= Cout) return;
  int co = n0 + lm;
  int K = Cin * 9;
  const float* wr = w + (i64)co * K;
  v8f acc = {};
  for (int k0 = 0; k0 < K; k0 += 4) {
    int ka = k0 + 2 * hi;
    int kb = ka + 1;
    int cia = ka / 9, ja = ka - cia * 9;
    int rya = ja / 3, rxa = ja - rya * 3;
    int cib = kb / 9, jb = kb - cib * 9;
    int ryb = jb / 3, rxb = jb - ryb * 3;
    v2f a, bb;
    a[0] = patch[cia * 54 + rya * 18 + lm + rxa];
    a[1] = patch[cib * 54 + ryb * 18 + lm + rxb];
    bb[0] = wr[ka];
    bb[1] = wr[kb];
    acc = __builtin_amdgcn_wmma_f32_16x16x4_f32(false, a, false, bb, (short)0,
                                                acc, false, false);
  }
  float bv = bias ? bias[co] : 0.f;
  i64 pix = (i64)y * W + x0;
  float* ob = out + (i64)b * out_bz + (i64)co * L + pix;
  const float* rb = R ? (R + (i64)b * R_bz + (i64)co * L + pix) : nullptr;
#pragma unroll
  for (int r = 0; r < 8; ++r) {
    int m = r + 8 * hi;
    float v = acc[r] + bv;
    if (rb) v += rb[m];
    ob[m] = v;          // contiguous 8 floats per lane
  }
}

// ---------------------------------------------------------------------------
// GroupNorm stats (one block per (batch,group)); channels within a group are
// contiguous in NCHW so the reduction is a contiguous streaming read.
// ---------------------------------------------------------------------------
__launch_bounds__(256)
__global__ void gn_stats(const float* __restrict__ in, float* __restrict__ stats,
                         int C, int G, int L)
{
  int bg = blockIdx.x;
  int b = bg / G, g = bg - b * G;
  int Cg = C / G;
  i64 base = (i64)b * C * L + (i64)g * Cg * L;
  int tot = Cg * L;
  float s = 0.f, ss = 0.f;
  for (int i = threadIdx.x; i < tot; i += 256) {
    float v = in[base + i];
    s += v; ss += v * v;
  }
  for (int o = 16; o; o >>= 1) { s += __shfl_xor(s, o, 32); ss += __shfl_xor(ss, o, 32); }
  __shared__ float sh[16];
  int wave = threadIdx.x >> 5, lane = threadIdx.x & 31;
  if (lane == 0) { sh[wave] = s; sh[wave + 8] = ss; }
  __syncthreads();
  if (threadIdx.x == 0) {
    float S = 0.f, SS = 0.f;
    for (int i = 0; i < 8; ++i) { S += sh[i]; SS += sh[i + 8]; }
    float mu = S / tot;
    float var = SS / tot - mu * mu;
    stats[bg * 2] = mu;
    stats[bg * 2 + 1] = rsqrtf(var + 1e-5f);
  }
}

__global__ void gn_apply_silu(const float* __restrict__ in,
                              const float* __restrict__ stats,
                              const float* __restrict__ gam,
                              const float* __restrict__ bet,
                              float* __restrict__ out,
                              int Btot, int C, int G, int L)
{
  int idx = blockIdx.x * blockDim.x + threadIdx.x;
  if (idx >= Btot) return;
  int cl = C * L;
  int b = idx / cl;
  int rem = idx - b * cl;
  int c = rem / L;
  int Cg = C / G;
  int g = c / Cg;
  float mu = stats[(b * G + g) * 2], rs = stats[(b * G + g) * 2 + 1];
  float v = (in[idx] - mu) * rs * gam[c] + bet[c];
  out[idx] = siluf(v);
}

// LayerNorm over last dim (C = 128 or 256), one wave per token.
__launch_bounds__(256)
__global__ void ln_kernel(const float* __restrict__ in, const float* __restrict__ g,
                          const float* __restrict__ b, float* __restrict__ out,
                          int T, int C)
{
  int wave = threadIdx.x >> 5, lane = threadIdx.x & 31;
  int t = blockIdx.x * 8 + wave;
  if (t >= T) return;
  const float* row = in + (i64)t * C;
  int E = C >> 5;
  float v[8];
  float s = 0.f;
  for (int e = 0; e < E; ++e) { v[e] = row[lane + (e << 5)]; s += v[e]; }
  for (int o = 16; o; o >>= 1) s += __shfl_xor(s, o, 32);
  float mu = s / C;
  float q = 0.f;
  for (int e = 0; e < E; ++e) { float d = v[e] - mu; q += d * d; }
  for (int o = 16; o; o >>= 1) q += __shfl_xor(q, o, 32);
  float rs = rsqrtf(q / C + 1e-5f);
  float* orow = out + (i64)t * C;
  for (int e = 0; e < E; ++e) {
    int c = lane + (e << 5);
    orow[c] = (v[e] - mu) * rs * g[c] + b[c];
  }
}

// Depthwise 3x3 + bias + SiLU, reading x1 slice of xz (NHWC, stride 2*Di).
__global__ void dwconv_silu(const float* __restrict__ xz, const float* __restrict__ w,
                            const float* __restrict__ bias, float* __restrict__ xc,
                            int Btot, int Di, int H, int W)
{
  int idx = blockIdx.x * blockDim.x + threadIdx.x;
  if (idx >= Btot) return;
  int L = H * W;
  int c = idx % Di;
  int l = (idx / Di) % L;
  int b = idx / (Di * L);
  int y = l / W, x = l - y * W;
  float acc = bias[c];
#pragma unroll
  for (int ry = 0; ry < 3; ++ry) {
    int gy = y + ry - 1;
    if (gy < 0 || gy >= H) continue;
#pragma unroll
    for (int rx = 0; rx < 3; ++rx) {
      int gx = x + rx - 1;
      if (gx < 0 || gx >= W) continue;
      acc += xz[((i64)b * L + gy * W + gx) * (2 * Di) + c] * w[c * 9 + ry * 3 + rx];
    }
  }
  xc[idx] = siluf(acc);
}

// Build the 4 directional scan copies, token-major: xs[b][k][l][d].
__global__ void make_xs(const float* __restrict__ xc, float* __restrict__ xs,
                        int Btot, int Di, int H, int W)
{
  int idx = blockIdx.x * blockDim.x + threadIdx.x;
  if (idx >= Btot) return;
  int L = H * W;
  int c = idx % Di;
  int l = (idx / Di) % L;
  int b = idx / (Di * L);
  float v = xc[idx];
  int y = l / W, x = l - y * W;
  int lt = x * H + y;
  i64 base = (i64)b * 4 * L * Di;
  xs[base + ((i64)0 * L + l) * Di + c] = v;
  xs[base + ((i64)1 * L + lt) * Di + c] = v;
  xs[base + ((i64)2 * L + (L - 1 - l)) * Di + c] = v;
  xs[base + ((i64)3 * L + (L - 1 - lt)) * Di + c] = v;
}

__global__ void neg_exp(const float* __restrict__ a, float* __restrict__ o, int n)
{
  int i = blockIdx.x * blockDim.x + threadIdx.x;
  if (i < n) o[i] = -__expf(a[i]);
}

// ---------------------------------------------------------------------------
// Chunked selective scan. h[n] <- exp(dt*A[n])*h[n] + dt*x*B[n]; y = <h,C> + D*x
// Pass1: per-(b,k,d,chunk) local scan from h=0 -> (P = exp(A*sum_dt), Q = h_end)
// Pass2: per-(b,k,d) sequential scan over 64 chunk summaries -> Hin per chunk
// Pass3: replay each chunk from Hin, emit y. B/C staged per-chunk in LDS.
// ---------------------------------------------------------------------------
#define NSTATE 16
#define CHUNK 64

__launch_bounds__(256)
__global__ void scan_pass1(const float* __restrict__ xs, const float* __restrict__ dts,
                           const float* __restrict__ xdbl, const float* __restrict__ Aneg,
                           float* __restrict__ P, float* __restrict__ Q,
                           int L, int Di)
{
  int chunk = blockIdx.x, k = blockIdx.y, b = blockIdx.z;
  int d = threadIdx.x;                 // blockDim.x == Di == 256
  int bk = b * 4 + k;
  int c0 = chunk * CHUNK;
  __shared__ float Bc[CHUNK * NSTATE];
  for (int idx = threadIdx.x; idx < CHUNK * NSTATE; idx += 256) {
    int i = idx >> 4, n = idx & 15;
    Bc[idx] = xdbl[((i64)bk * L + c0 + i) * 40 + 8 + n];
  }
  __syncthreads();
  float A[NSTATE], h[NSTATE];
  const float* Ad = Aneg + ((i64)k * Di + d) * NSTATE;
#pragma unroll
  for (int n = 0; n < NSTATE; ++n) { A[n] = Ad[n]; h[n] = 0.f; }
  float sumdt = 0.f;
  const float* xrow = xs + (i64)bk * L * Di;
  const float* drow = dts + (i64)bk * L * Di;
  for (int i = 0; i < CHUNK; ++i) {
    i64 e = ((i64)(c0 + i)) * Di + d;
    float dt = drow[e];
    float xv = xrow[e];
    float dtx = dt * xv;
    sumdt += dt;
#pragma unroll
    for (int n = 0; n < NSTATE; ++n) {
      float dA = __expf(dt * A[n]);
      h[n] = dA * h[n] + dtx * Bc[i * NSTATE + n];
    }
  }
  i64 ob = (((i64)bk * Di + d) * (L / CHUNK) + chunk) * NSTATE;
#pragma unroll
  for (int n = 0; n < NSTATE; ++n) {
    P[ob + n] = __expf(A[n] * sumdt);   // exact product of per-step exp(dt*A)
    Q[ob + n] = h[n];
  }
}

__global__ void scan_pass2(const float* __restrict__ P, const float* __restrict__ Q,
                           float* __restrict__ Hin, int nbkd, int nchunks)
{
  int t = blockIdx.x * blockDim.x + threadIdx.x;
  if (t >= nbkd) return;
  i64 base = (i64)t * nchunks * NSTATE;
  float h[NSTATE];
#pragma unroll
  for (int n = 0; n < NSTATE; ++n) h[n] = 0.f;
  for (int c = 0; c < nchunks; ++c) {
    i64 e = base + (i64)c * NSTATE;
#pragma unroll
    for (int n = 0; n < NSTATE; ++n) {
      Hin[e + n] = h[n];
      h[n] = P[e + n] * h[n] + Q[e + n];
    }
  }
}

__launch_bounds__(256)
__global__ void scan_pass3(const float* __restrict__ xs, const float* __restrict__ dts,
                           const float* __restrict__ xdbl, const float* __restrict__ Aneg,
                           const float* __restrict__ Hin, const float* __restrict__ Ds,
                           float* __restrict__ ys, int L, int Di)
{
  int chunk = blockIdx.x, k = blockIdx.y, b = blockIdx.z;
  int d = threadIdx.x;
  int bk = b * 4 + k;
  int c0 = chunk * CHUNK;
  __shared__ float Bc[CHUNK * NSTATE];
  __shared__ float Cc[CHUNK * NSTATE];
  for (int idx = threadIdx.x; idx < CHUNK * NSTATE; idx += 256) {
    int i = idx >> 4, n = idx & 15;
    i64 rb = ((i64)bk * L + c0 + i) * 40;
    Bc[idx] = xdbl[rb + 8 + n];
    Cc[idx] = xdbl[rb + 24 + n];
  }
  __syncthreads();
  float A[NSTATE], h[NSTATE];
  const float* Ad = Aneg + ((i64)k * Di + d) * NSTATE;
  i64 hb = (((i64)bk * Di + d) * (L / CHUNK) + chunk) * NSTATE;
#pragma unroll
  for (int n = 0; n < NSTATE; ++n) { A[n] = Ad[n]; h[n] = Hin[hb + n]; }
  float Dv = Ds[k * Di + d];
  const float* xrow = xs + (i64)bk * L * Di;
  const float* drow = dts + (i64)bk * L * Di;
  float* yrow = ys + (i64)bk * L * Di;
  for (int i = 0; i < CHUNK; ++i) {
    i64 e = ((i64)(c0 + i)) * Di + d;
    float dt = drow[e];
    float xv = xrow[e];
    float dtx = dt * xv;
    float y = 0.f;
#pragma unroll
    for (int n = 0; n < NSTATE; ++n) {
      float dA = __expf(dt * A[n]);
      h[n] = dA * h[n] + dtx * Bc[i * NSTATE + n];
      y += h[n] * Cc[i * NSTATE + n];
    }
    yrow[e] = y + Dv * xv;    // ys may alias xs: element read-before-write
  }
}

// Merge the 4 directions + LayerNorm(out_norm) + gate by silu(z). Wave/token.
__launch_bounds__(256)
__global__ void merge_ln_gate(const float* __restrict__ ys, const float* __restrict__ xz,
                              const float* __restrict__ g, const float* __restrict__ bta,
                              float* __restrict__ yln, int Bsz, int H, int W, int Di)
{
  int wave = threadIdx.x >> 5, lane = threadIdx.x & 31;
  int L = H * W;
  int t = blockIdx.x * 8 + wave;
  if (t >= Bsz * L) return;
  int b = t / L, l = t - b * L;
  int y = l / W, x = l - y * W;
  int lt = x * H + y;
  i64 base = (i64)b * 4 * L * Di;
  const float* r0 = ys + base + ((i64)0 * L + l) * Di;
  const float* r1 = ys + base + ((i64)1 * L + lt) * Di;
  const float* r2 = ys + base + ((i64)2 * L + (L - 1 - l)) * Di;
  const float* r3 = ys + base + ((i64)3 * L + (L - 1 - lt)) * Di;
  int E = Di >> 5;   // 8
  float v[8];
  float s = 0.f;
  for (int e = 0; e < E; ++e) {
    int c = lane + (e << 5);
    float m = r0[c] + r1[c] + r2[c] + r3[c];
    v[e] = m; s += m;
  }
  for (int o = 16; o; o >>= 1) s += __shfl_xor(s, o, 32);
  float mu = s / Di;
  float q = 0.f;
  for (int e = 0; e < E; ++e) { float dd = v[e] - mu; q += dd * dd; }
  for (int o = 16; o; o >>= 1) q += __shfl_xor(q, o, 32);
  float rs = rsqrtf(q / Di + 1e-5f);
  const float* zrow = xz + (i64)t * (2 * Di) + Di;
  float* orow = yln + (i64)t * Di;
  for (int e = 0; e < E; ++e) {
    int c = lane + (e << 5);
    float nv = (v[e] - mu) * rs * g[c] + bta[c];
    orow[c] = nv * siluf(zrow[c]);
  }
}

// Misc data movement
__global__ void upsample2x(const float* __restrict__ x, float* __restrict__ xu,
                           int Btot, int C)
{
  int idx = blockIdx.x * blockDim.x + threadIdx.x;
  if (idx >= Btot) return;
  int x2 = idx & 63;
  int y2 = (idx >> 6) & 63;
  int c = (idx >> 12) % C;
  int b = idx / (C * 4096);
  xu[idx] = x[((i64)(b * C + c) * 32 + (y2 >> 1)) * 32 + (x2 >> 1)];
}

__global__ void copy_skip(const float* __restrict__ skip, float* __restrict__ cat,
                          int Btot)
{
  int idx = blockIdx.x * blockDim.x + threadIdx.x;
  if (idx >= Btot) return;
  int b = idx / (128 * 4096);
  int rem = idx - b * 128 * 4096;
  cat[(i64)b * 256 * 4096 + 128 * 4096 + rem] = skip[idx];
}

__global__ void nchw_to_nhwc(const float* __restrict__ in, float* __restrict__ out,
                             int Btot, int C, int L)
{
  int idx = blockIdx.x * blockDim.x + threadIdx.x;
  if (idx >= Btot) return;
  int c = idx % C;
  int l = (idx / C) % L;
  int b = idx / (C * L);
  out[idx] = in[((i64)b * C + c) * L + l];
}

__global__ void nhwc_to_nchw(const float* __restrict__ in, float* __restrict__ out,
                             int Btot, int C, int L)
{
  int idx = blockIdx.x * blockDim.x + threadIdx.x;
  if (idx >= Btot) return;
  int l = idx % L;
  int c = (idx / L) % C;
  int b = idx / (C * L);
  out[idx] = in[((i64)b * L + l) * C + c];
}

// ---------------------------------------------------------------------------
// Host-side orchestration
// ---------------------------------------------------------------------------
static inline dim3 g1(i64 n) { return dim3((unsigned)((n + 255) / 256)); }

static void launch_gemm(hipStream_t s, const float* A, const float* B,
                        const float* bias, const float* R, float* C,
                        int M, int N, int K, int act, int Z, int zdiv,
                        i64 A_zb, i64 A_zk, i64 As_m, i64 As_k,
                        i64 B_zk, i64 Bs_n, i64 Bs_k,
                        i64 C_zb, i64 C_zk, i64 Cs_m, i64 Cs_n,
                        i64 bias_zk, i64 R_zb, i64 R_zk)
{
  dim3 grid(M / 16, (N + 127) / 128, Z);
  gemm_wmma<<<grid, 256, 0, s>>>(A, B, bias, R, C, M, N, K, act, zdiv,
                                 A_zb, A_zk, As_m, As_k, B_zk, Bs_n, Bs_k,
                                 C_zb, C_zk, Cs_m, Cs_n, bias_zk, R_zb, R_zk);
}

static void run_res_block(hipStream_t s, const float* const* P, int p0, bool sc,
                          const float* in, float* out, int Cin, int Cout,
                          float* tA, float* tB, float* c1, float* scb,
                          float* stats, int Bsz, int L)
{
  const int G = 8;
  gn_stats<<<dim3(Bsz * G), 256, 0, s>>>(in, stats, Cin, G, L);
  i64 tot = (i64)Bsz * Cin * L;
  gn_apply_silu<<<g1(tot), 256, 0, s>>>(in, stats, P[p0 + 0], P[p0 + 1], tA,
                                        (int)tot, Cin, G, L);
  conv3x3_wmma<<<dim3(256, (Cout + 127) / 128, Bsz), 256, 0, s>>>(
      tA, P[p0 + 2], P[p0 + 3], nullptr, c1, Cin, Cout, 64, 64,
      (i64)Cin * L, (i64)Cout * L, 0);
  gn_stats<<<dim3(Bsz * G), 256, 0, s>>>(c1, stats + 64, Cout, G, L);
  i64 tot2 = (i64)Bsz * Cout * L;
  gn_apply_silu<<<g1(tot2), 256, 0, s>>>(c1, stats + 64, P[p0 + 4], P[p0 + 5], tB,
                                         (int)tot2, Cout, G, L);
  const float* resid = in;
  i64 Rbz = (i64)Cin * L;
  if (sc) {
    // shortcut: reuse input stats (mean/var independent of affine params)
    gn_apply_silu<<<g1(tot), 256, 0, s>>>(in, stats, P[p0 + 8], P[p0 + 9], tA,
                                          (int)tot, Cin, G, L);
    launch_gemm(s, tA, P[p0 + 10], P[p0 + 11], nullptr, scb, L, Cout, Cin, 0,
                Bsz, 1,
                (i64)Cin * L, 0, 1, L,         // A (NCHW: m=pixel, k=channel)
                0, Cin, 1,                     // B = sc_w (Cout, Cin)
                (i64)Cout * L, 0, 1, L,        // C (NCHW)
                0, 0, 0);
    resid = scb;
    Rbz = (i64)Cout * L;
  }
  conv3x3_wmma<<<dim3(256, (Cout + 127) / 128, Bsz), 256, 0, s>>>(
      tB, P[p0 + 6], P[p0 + 7], resid, out, Cout, Cout, 64, 64,
      (i64)Cout * L, (i64)Cout * L, Rbz);
}

static void run_vss(hipStream_t s, const float* const* P, int p0, float* hn,
                    float* tln, float* xz, float* xc, float* xs, float* xdbl,
                    float* dts, float* aneg, float* Pb, float* Qb, float* Hb,
                    float* yln, float* mbuf, int Bsz)
{
  const int L = 4096, d = 128, Di = 256, Rr = 8, Ns = 16, Kd = 4;
  int T = Bsz * L;
  // ln1
  ln_kernel<<<dim3((T + 7) / 8), 256, 0, s>>>(hn, P[p0 + 0], P[p0 + 1], tln, T, d);
  // in_proj: (T,128) @ (512,128)^T -> xz (T,512)
  launch_gemm(s, tln, P[p0 + 2], nullptr, nullptr, xz, T, 2 * Di, d, 0, 1, 1,
              0, 0, d, 1, 0, d, 1, 0, 0, 2 * Di, 1, 0, 0, 0);
  // depthwise conv + silu on x1 slice of xz
  dwconv_silu<<<g1((i64)T * Di), 256, 0, s>>>(xz, P[p0 + 3], P[p0 + 4], xc,
                                              T * Di, Di, 64, 64);
  // 4 directional copies, token-major
  make_xs<<<g1((i64)T * Di), 256, 0, s>>>(xc, xs, T * Di, Di, 64, 64);
  // x_dbl: per (b,k): (L,256) @ (40,256)^T -> (L,40)
  launch_gemm(s, xs, P[p0 + 5], nullptr, nullptr, xdbl, L, 40, Di, 0, Bsz * Kd, Kd,
              (i64)Kd * L * Di, (i64)L * Di, Di, 1,
              (i64)40 * Di, Di, 1,
              (i64)Kd * L * 40, (i64)L * 40, 40, 1,
              0, 0, 0);
  // dts: per (b,k): (L,8) @ (256,8)^T + dt_b -> softplus -> (L,256)
  launch_gemm(s, xdbl, P[p0 + 6], P[p0 + 7], nullptr, dts, L, Di, Rr, 3, Bsz * Kd, Kd,
              (i64)Kd * L * 40, (i64)L * 40, 40, 1,
              (i64)Di * Rr, Rr, 1,
              (i64)Kd * L * Di, (i64)L * Di, Di, 1,
              Di, 0, 0);
  // A = -exp(A_logs)
  neg_exp<<<g1(Kd * Di * Ns), 256, 0, s>>>(P[p0 + 8], aneg, Kd * Di * Ns);
  // chunked linear scan (ys aliases xs)
  scan_pass1<<<dim3(L / CHUNK, Kd, Bsz), 256, 0, s>>>(xs, dts, xdbl, aneg, Pb, Qb, L, Di);
  scan_pass2<<<g1(Bsz * Kd * Di), 256, 0, s>>>(Pb, Qb, Hb, Bsz * Kd * Di, L / CHUNK);
  scan_pass3<<<dim3(L / CHUNK, Kd, Bsz), 256, 0, s>>>(xs, dts, xdbl, aneg, Hb,
                                                      P[p0 + 9], xs, L, Di);
  // merge + out_norm LN + silu(z) gate
  merge_ln_gate<<<dim3((T + 7) / 8), 256, 0, s>>>(xs, xz, P[p0 + 10], P[p0 + 11],
                                                  yln, Bsz, 64, 64, Di);
  // out_proj + residual (in place on hn; per-element read-before-write)
  launch_gemm(s, yln, P[p0 + 12], nullptr, hn, hn, T, d, Di, 0, 1, 1,
              0, 0, Di, 1, 0, Di, 1, 0, 0, d, 1, 0, 0, 0);
  // MLP
  ln_kernel<<<dim3((T + 7) / 8), 256, 0, s>>>(hn, P[p0 + 13], P[p0 + 14], tln, T, d);
  launch_gemm(s, tln, P[p0 + 15], P[p0 + 16], nullptr, mbuf, T, 4 * d, d, 2, 1, 1,
              0, 0, d, 1, 0, d, 1, 0, 0, 4 * d, 1, 0, 0, 0);
  launch_gemm(s, mbuf, P[p0 + 17], P[p0 + 18], hn, hn, T, d, 4 * d, 0, 1, 1,
              0, 0, 4 * d, 1, 0, 4 * d, 1, 0, 0, d, 1, 0, 0, 0);
}

extern "C" void kernel_launch(void* const* d_in, const int* in_sizes, int n_in,
                              void* d_out, int out_size, void* d_ws, size_t ws_size,
                              hipStream_t stream)
{
  (void)in_sizes; (void)out_size; (void)ws_size;
  const float* P[64];
  for (int i = 0; i < n_in && i < 64; ++i) P[i] = (const float*)d_in[i];
  const float* x = P[0];
  const float* skip = P[1];
  float* out = (float*)d_out;
  float* ws = (float*)d_ws;

  const int Bsz = 2, L = 4096;
  size_t off = 0;
  auto alloc = [&](size_t n) { float* p = ws + off; off += n; return p; };

  float* xu   = alloc((size_t)Bsz * 256 * L);  // upsampled input; reused as tA
  float* cat  = alloc((size_t)Bsz * 256 * L);  // concat buffer; reused as yln
  float* tB   = alloc((size_t)Bsz * 128 * L);
  float* c1   = alloc((size_t)Bsz * 128 * L);
  float* scb  = alloc((size_t)Bsz * 128 * L);
  float* h    = alloc((size_t)Bsz * 128 * L);
  float* hn   = alloc((size_t)Bsz * L * 128);
  float* tln  = alloc((size_t)Bsz * L * 128);
  float* xz   = alloc((size_t)Bsz * L * 512);
  float* xc   = alloc((size_t)Bsz * L * 256);
  float* xs   = alloc((size_t)Bsz * 4 * L * 256);  // also holds ys (aliased)
  float* xdbl = alloc((size_t)Bsz * 4 * L * 40);
  float* dts  = alloc((size_t)Bsz * 4 * L * 256);  // also reused as MLP hidden
  float* aneg = alloc((size_t)4 * 256 * 16);
  float* Pb   = alloc((size_t)Bsz * 4 * 256 * (L / CHUNK) * NSTATE);
  float* Qb   = alloc((size_t)Bsz * 4 * 256 * (L / CHUNK) * NSTATE);
  float* Hb   = alloc((size_t)Bsz * 4 * 256 * (L / CHUNK) * NSTATE);
  float* stats = alloc(256);
  float* tA   = xu;     // reuse: xu dead after up_conv
  float* yln  = cat;    // reuse: cat dead after first res_block
  float* mbuf = dts;    // reuse: dts dead after scan pass3

  // 1. nearest-upsample 2x
  upsample2x<<<g1((i64)Bsz * 256 * L), 256, 0, stream>>>(x, xu, Bsz * 256 * L, 256);
  // 2. up_conv 3x3 (256->128) into cat channels [0,128)
  conv3x3_wmma<<<dim3(256, 1, Bsz), 256, 0, stream>>>(
      xu, P[2], P[3], nullptr, cat, 256, 128, 64, 64,
      (i64)256 * L, (i64)256 * L, 0);
  // 3. skip into cat channels [128,256)
  copy_skip<<<g1((i64)Bsz * 128 * L), 256, 0, stream>>>(skip, cat, Bsz * 128 * L);
  // 4. in_res: 256->128 with 1x1 shortcut
  run_res_block(stream, P, 4, true, cat, h, 256, 128, tA, tB, c1, scb, stats, Bsz, L);
  // 5. NCHW -> NHWC for VSS blocks
  nchw_to_nhwc<<<g1((i64)Bsz * 128 * L), 256, 0, stream>>>(h, hn, Bsz * 128 * L, 128, L);
  // 6. two VSS blocks (params at 16 and 35)
  run_vss(stream, P, 16, hn, tln, xz, xc, xs, xdbl, dts, aneg, Pb, Qb, Hb, yln, mbuf, Bsz);
  run_vss(stream, P, 35, hn, tln, xz, xc, xs, xdbl, dts, aneg, Pb, Qb, Hb, yln, mbuf, Bsz);
  // 7. back to NCHW, final res block (128->128, identity shortcut) -> d_out
  nhwc_to_nchw<<<g1((i64)Bsz * 128 * L), 256, 0, stream>>>(hn, scb, Bsz * 128 * L, 128, L);
  run_res_block(stream, P, 54, false, scb, out, 128, 128, tA, tB, c1, nullptr, stats, Bsz, L);
}